// NTN_33122787786769
// MI455X (gfx1250) — compile-verified
//
#include <hip/hip_runtime.h>
#include <hip/hip_bf16.h>
#include <math.h>

typedef __attribute__((ext_vector_type(2))) float v2f;
typedef __attribute__((ext_vector_type(8))) float v8f;

#define BATCH   1024
#define EMB     128
#define TD      10
#define NREL    50
#define NITEMS  2048
#define MAX_TILES 178   // worst case: 128 full tiles + 50 partial tails

// ---- workspace layout (int region then float region) ----
#define WS_NTILES 0
#define WS_OFFS   1            // 51 ints
#define WS_TREL   64           // 178 ints
#define WS_TBASE  256          // 178 ints
#define WS_TEND   448          // 178 ints
#define WS_BUCKET 640          // 2048 ints
#define WS_SCORES 4096         // 2048 floats (offset in 32-bit words)

// ------------------------------------------------------------------
// Kernel 1: bucket the 2048 (pos+neg) items by relation, build tile list
// ------------------------------------------------------------------
__global__ __launch_bounds__(1024)
void ntn_bucket(const int* __restrict__ pos, const int* __restrict__ neg,
                int* __restrict__ ws) {
    __shared__ int cnt[NREL];
    __shared__ int offs[NREL + 1];
    __shared__ int cur[NREL];
    int tid = threadIdx.x;
    if (tid < NREL) cnt[tid] = 0;
    __syncthreads();

    for (int item = tid; item < NITEMS; item += 1024) {
        int row = item & (BATCH - 1);
        const int* trip = (item < BATCH) ? pos : neg;
        int r = trip[row * 3 + 1];
        atomicAdd(&cnt[r], 1);
    }
    __syncthreads();

    if (tid == 0) {
        int off = 0;
        for (int r = 0; r < NREL; ++r) { offs[r] = off; cur[r] = off; off += cnt[r]; }
        offs[NREL] = off;
        int nt = 0;
        for (int r = 0; r < NREL; ++r) {
            for (int b = offs[r]; b < offs[r + 1]; b += 16) {
                ws[WS_TREL  + nt] = r;
                ws[WS_TBASE + nt] = b;
                ws[WS_TEND  + nt] = offs[r + 1];
                ++nt;
            }
        }
        ws[WS_NTILES] = nt;
        for (int r = 0; r <= NREL; ++r) ws[WS_OFFS + r] = offs[r];
    }
    __syncthreads();

    for (int item = tid; item < NITEMS; item += 1024) {
        int row = item & (BATCH - 1);
        const int* trip = (item < BATCH) ? pos : neg;
        int r = trip[row * 3 + 1];
        int slot = atomicAdd(&cur[r], 1);
        ws[WS_BUCKET + slot] = item;
    }
}

// ------------------------------------------------------------------
// Kernel 2: per (relation, 16-column) tile, WMMA fp32 bilinear + rest
//   Y = W[r,k] (128x128)  x  T^T (128x16)  via v_wmma_f32_16x16x4_f32,
//   bilinear[k,c] = h_c . Y[:,c], then linear + tanh + u-weighted sum.
//   4 waves per block split the 10 tensor slices; h/t tiles live in LDS.
// ------------------------------------------------------------------
__global__ __launch_bounds__(128)
void ntn_main(const int* __restrict__ pos, const int* __restrict__ neg,
              const float* __restrict__ emb,
              const float* __restrict__ rel_W, const float* __restrict__ rel_b,
              const float* __restrict__ rel_V, const float* __restrict__ rel_u,
              const int* __restrict__ ws, float* __restrict__ scores) {
    __shared__ float Hs[16][132];     // padded stride: avoid 64-bank conflicts
    __shared__ float Ts[16][132];
    __shared__ float BIL[TD][16];
    __shared__ int hentS[16], tentS[16], itemS[16];

    int nt = ws[WS_NTILES];
    if (blockIdx.x >= nt) return;
    int r    = ws[WS_TREL  + blockIdx.x];
    int base = ws[WS_TBASE + blockIdx.x];
    int end  = ws[WS_TEND  + blockIdx.x];

    int tid = threadIdx.x;
    if (tid < 16) {
        int slot = base + tid;
        int item = -1, he = 0, te = 0;
        if (slot < end) {
            item = ws[WS_BUCKET + slot];
            int row = item & (BATCH - 1);
            const int* trip = (item < BATCH) ? pos : neg;
            he = trip[row * 3 + 0];
            te = trip[row * 3 + 2];
        }
        itemS[tid] = item; hentS[tid] = he; tentS[tid] = te;
    }
    __syncthreads();

    for (int idx = tid; idx < 16 * EMB; idx += 128) {
        int c = idx >> 7, d = idx & 127;
        bool valid = itemS[c] >= 0;
        Hs[c][d] = valid ? emb[(size_t)hentS[c] * EMB + d] : 0.f;
        Ts[c][d] = valid ? emb[(size_t)tentS[c] * EMB + d] : 0.f;
    }
    __syncthreads();

    int lane = tid & 31;
    int wv   = tid >> 5;
    int m    = lane & 15;          // A: row M / B: col N handled by this lane
    int koff = (lane >> 4) * 2;    // K sub-offset per lane half

    // each wave owns k = wv, wv+4, wv+8  (uniform per wave -> EXEC all ones)
    for (int k = wv; k < TD; k += 4) {
        const float* Wk = rel_W + ((size_t)(r * TD + k)) * EMB * EMB;
        float part = 0.f;
        for (int it = 0; it < 8; ++it) {
            int i0 = it * 16;
            const float* Wrow = Wk + (size_t)(i0 + m) * EMB + koff;
            v8f acc = {};
            #pragma unroll 8
            for (int jc = 0; jc < 32; ++jc) {
                int j0 = jc * 4;
                v2f a, b;
                a.x = Wrow[j0];            // W[r,k, i0+m, j0+koff]
                a.y = Wrow[j0 + 1];        // contiguous pair -> b64 load
                b.x = Ts[m][j0 + koff];    // B[K,N] = t_N[j]
                b.y = Ts[m][j0 + koff + 1];
                acc = __builtin_amdgcn_wmma_f32_16x16x4_f32(
                        false, a, false, b, (short)0, acc, false, false);
            }
            // C layout: lane l, vgpr v holds Y[i0 + v + 8*(l>>4), n = l&15]
            int ibase = i0 + 8 * (lane >> 4);
            #pragma unroll
            for (int v = 0; v < 8; ++v)
                part += Hs[m][ibase + v] * acc[v];
        }
        part += __shfl_xor(part, 16);      // fold two i-halves (wave32)
        if (lane < 16) BIL[k][lane] = part;
    }
    __syncthreads();

    if (tid < 16 && itemS[tid] >= 0) {
        int c = tid;
        float s = 0.f;
        for (int k = 0; k < TD; ++k) {
            const float* V = rel_V + (size_t)(r * TD + k) * 2 * EMB;
            float lv = 0.f;
            for (int i = 0; i < EMB; ++i)
                lv += V[i] * Hs[c][i] + V[EMB + i] * Ts[c][i];
            float act = tanhf(BIL[k][c] + lv + rel_b[r * TD + k]);
            s += rel_u[r * TD + k] * act;
        }
        scores[itemS[c]] = s;
    }
}

// ------------------------------------------------------------------
// Kernel 3: mean hinge loss over the batch
// ------------------------------------------------------------------
__global__ __launch_bounds__(256)
void ntn_loss(const float* __restrict__ scores, float* __restrict__ out) {
    __shared__ float red[256];
    int tid = threadIdx.x;
    float s = 0.f;
    for (int b = tid; b < BATCH; b += 256) {
        float p = scores[b];
        float n = scores[BATCH + b];
        s += fmaxf(n - p + 1.0f, 0.f);
    }
    red[tid] = s;
    __syncthreads();
    for (int st = 128; st > 0; st >>= 1) {
        if (tid < st) red[tid] += red[tid + st];
        __syncthreads();
    }
    if (tid == 0) out[0] = red[0] / (float)BATCH;
}

extern "C" void kernel_launch(void* const* d_in, const int* in_sizes, int n_in,
                              void* d_out, int out_size, void* d_ws, size_t ws_size,
                              hipStream_t stream) {
    const int*   pos   = (const int*)  d_in[0];
    const int*   neg   = (const int*)  d_in[1];
    const float* emb   = (const float*)d_in[2];
    const float* rel_W = (const float*)d_in[3];
    const float* rel_b = (const float*)d_in[4];
    const float* rel_V = (const float*)d_in[5];
    const float* rel_u = (const float*)d_in[6];

    int*   ws_i   = (int*)d_ws;
    float* scores = (float*)d_ws + WS_SCORES;
    float* out    = (float*)d_out;

    ntn_bucket<<<1, 1024, 0, stream>>>(pos, neg, ws_i);
    ntn_main<<<MAX_TILES, 128, 0, stream>>>(pos, neg, emb, rel_W, rel_b, rel_V,
                                            rel_u, ws_i, scores);
    ntn_loss<<<1, 256, 0, stream>>>(scores, out);
}